// DifferentiableDistanceGeometry_64321430225397
// MI455X (gfx1250) — compile-verified
//
#include <hip/hip_runtime.h>
#include <cmath>

#define L 4096
#define LL (4096ull * 4096ull)
#define POWER_ITERS 96
#define NUM_ITERS 50

typedef float v2f __attribute__((ext_vector_type(2)));
typedef float v8f __attribute__((ext_vector_type(8)));

// ---------------------------------------------------------------------------
// 1) Symmetrize inputs: pred_d = (P+P^T)/2, weight = (C+C^T)/2 * mask
// ---------------------------------------------------------------------------
__global__ void sym_kernel(const float* __restrict__ P, const float* __restrict__ C,
                           const float* __restrict__ M,
                           float* __restrict__ predd, float* __restrict__ wght) {
    size_t idx = (size_t)blockIdx.x * blockDim.x + threadIdx.x;
    if (idx >= LL) return;
    size_t i = idx >> 12, j = idx & 4095;
    size_t t = j * (size_t)L + i;
    predd[idx] = 0.5f * (P[idx] + P[t]);
    wght[idx]  = 0.5f * (C[idx] + C[t]) * M[idx];
}

// ---------------------------------------------------------------------------
// 2) Row reductions for double-centering: rowmean of D^2, rowsum of weight
// ---------------------------------------------------------------------------
__global__ __launch_bounds__(256) void rowsum_kernel(const float* __restrict__ predd,
                                                     const float* __restrict__ wght,
                                                     float* __restrict__ rmean,
                                                     float* __restrict__ rsumw) {
    __shared__ float s1[256], s2[256];
    int row = blockIdx.x, tid = threadIdx.x;
    const float* p = predd + (size_t)row * L;
    const float* w = wght  + (size_t)row * L;
    float sd = 0.f, sw = 0.f;
    for (int j = tid; j < L; j += 256) { float pv = p[j]; sd += pv * pv; sw += w[j]; }
    s1[tid] = sd; s2[tid] = sw; __syncthreads();
    for (int s = 128; s > 0; s >>= 1) {
        if (tid < s) { s1[tid] += s1[tid + s]; s2[tid] += s2[tid + s]; }
        __syncthreads();
    }
    if (tid == 0) { rmean[row] = s1[0] * (1.0f / L); rsumw[row] = s2[0]; }
}

// scal[0] = grandmean of D^2, scal[1] = sum(weight)
__global__ __launch_bounds__(256) void scalars_kernel(const float* __restrict__ rmean,
                                                      const float* __restrict__ rsumw,
                                                      float* __restrict__ scal) {
    __shared__ float s1[256], s2[256];
    int tid = threadIdx.x;
    float a = 0.f, b = 0.f;
    for (int i = tid; i < L; i += 256) { a += rmean[i]; b += rsumw[i]; }
    s1[tid] = a; s2[tid] = b; __syncthreads();
    for (int s = 128; s > 0; s >>= 1) {
        if (tid < s) { s1[tid] += s1[tid + s]; s2[tid] += s2[tid + s]; }
        __syncthreads();
    }
    if (tid == 0) { scal[0] = s1[0] * (1.0f / L); scal[1] = s2[0]; }
}

// ---------------------------------------------------------------------------
// 3) Deterministic pseudo-random init of X (4096 x 16, all 16 columns active)
// ---------------------------------------------------------------------------
__global__ void initX_kernel(float* __restrict__ X) {
    int idx = blockIdx.x * blockDim.x + threadIdx.x;
    if (idx >= L * 16) return;
    int i = idx >> 4, c = idx & 15;
    unsigned h = (unsigned)i * 2654435761u ^ ((unsigned)c * 0x9E3779B9u + 0x7F4A7C15u);
    h ^= h >> 13; h *= 0x85EBCA6Bu; h ^= h >> 16;
    X[idx] = (float)(h & 0xFFFFFF) * (1.0f / 16777216.0f) - 0.5f;
}

__global__ void zero_kernel(float* __restrict__ p, int n) {
    int i = blockIdx.x * blockDim.x + threadIdx.x;
    if (i < n) p[i] = 0.f;
}

// ---------------------------------------------------------------------------
// 4) Subspace-iteration GEMM: Y = B @ X  via V_WMMA_F32_16X16X4_F32.
//    B_ij = -0.5*(pred_d_ij^2 - rmean_i - rmean_j + g), computed on the fly.
//    X is 4096x16 (panel == WMMA N). One wave per 16-row tile; EXEC all ones.
//    A frag (16x4 f32): lane = M row (lane&15), lane[4] selects K pair.
// ---------------------------------------------------------------------------
__global__ __launch_bounds__(32) void gemm_power_kernel(const float* __restrict__ predd,
                                                        const float* __restrict__ rmean,
                                                        const float* __restrict__ scal,
                                                        const float* __restrict__ X,
                                                        float* __restrict__ Y) {
    const int m0   = blockIdx.x * 16;
    const int lane = threadIdx.x;          // 0..31, full wave
    const int mr   = lane & 15;
    const int ko   = (lane >> 4) << 1;     // 0 or 2
    const float g  = scal[0];
    const int row  = m0 + mr;
    const float ri = rmean[row];
    const float* prow = predd + (size_t)row * L;

    v8f acc = {};
    for (int k = 0; k < L; k += 16) {
        __builtin_prefetch(prow + k + 2048, 0, 0);   // global_prefetch_b8, 8KB ahead
#pragma unroll
        for (int kk = k; kk < k + 16; kk += 4) {
            v2f p = *(const v2f*)(prow + kk + ko);
            v2f a;
            a.x = -0.5f * (p.x * p.x - ri - rmean[kk + ko]     + g);
            a.y = -0.5f * (p.y * p.y - ri - rmean[kk + ko + 1] + g);
            v2f b;
            b.x = X[(kk + ko)     * 16 + mr];
            b.y = X[(kk + ko + 1) * 16 + mr];
            acc = __builtin_amdgcn_wmma_f32_16x16x4_f32(false, a, false, b,
                                                        (short)0, acc, false, false);
        }
    }
    // C/D layout: VGPR r -> M = r + 8*lane[4], N = lane&15
    const int nc    = lane & 15;
    const int rbase = (lane >> 4) * 8;
#pragma unroll
    for (int r = 0; r < 8; ++r) {
        int orow = m0 + rbase + r;
        Y[orow * 16 + nc] = acc[r];
    }
}

// ---------------------------------------------------------------------------
// 5) 16x16 Gram via WMMA: part_b = (A^T B) over K-slab b of 256 rows.
//    For a column panel, the A^T-fragment and B-fragment loads are identical:
//    lane holds element [(k+ko)*16 + lane&15].
// ---------------------------------------------------------------------------
__global__ __launch_bounds__(32) void gram_wmma_kernel(const float* __restrict__ A,
                                                       const float* __restrict__ Bm,
                                                       float* __restrict__ part) {
    const int lane = threadIdx.x;
    const int mr   = lane & 15;
    const int ko   = (lane >> 4) << 1;
    const int k0   = blockIdx.x * 256;
    v8f acc = {};
    for (int k = k0; k < k0 + 256; k += 4) {
        v2f a, b;
        a.x = A[(k + ko)     * 16 + mr];
        a.y = A[(k + ko + 1) * 16 + mr];
        b.x = Bm[(k + ko)     * 16 + mr];
        b.y = Bm[(k + ko + 1) * 16 + mr];
        acc = __builtin_amdgcn_wmma_f32_16x16x4_f32(false, a, false, b,
                                                    (short)0, acc, false, false);
    }
    const int nc    = lane & 15;
    const int rbase = (lane >> 4) * 8;
    float* out = part + blockIdx.x * 256;
#pragma unroll
    for (int r = 0; r < 8; ++r) out[(rbase + r) * 16 + nc] = acc[r];
}

// Sum the 16 partial 16x16 Grams -> G[256]
__global__ void gram_reduce_kernel(const float* __restrict__ part, float* __restrict__ G) {
    int t = threadIdx.x;      // 256 threads, one Gram entry each
    float s = 0.f;
    for (int b = 0; b < 16; ++b) s += part[b * 256 + t];
    G[t] = s;
}

// ---------------------------------------------------------------------------
// 6) CholeskyQR over all 16 columns: R = chol(G) built once per block in LDS,
//    then each thread solves its row: X_row = Y_row * R^-1.
// ---------------------------------------------------------------------------
__global__ __launch_bounds__(256) void cholqr16_kernel(const float* __restrict__ Y,
                                                       const float* __restrict__ G,
                                                       float* __restrict__ X) {
    __shared__ float R[256];
    int tid = threadIdx.x;
    if (tid == 0) {
        for (int i = 0; i < 16; ++i) {
            float d = G[i * 16 + i];
            for (int k = 0; k < i; ++k) d -= R[k * 16 + i] * R[k * 16 + i];
            float rii = sqrtf(fmaxf(d, 1e-20f));
            R[i * 16 + i] = rii;
            float inv = 1.0f / rii;
            for (int j = i + 1; j < 16; ++j) {
                float s = G[i * 16 + j];
                for (int k = 0; k < i; ++k) s -= R[k * 16 + i] * R[k * 16 + j];
                R[i * 16 + j] = s * inv;
            }
        }
    }
    __syncthreads();
    int i = blockIdx.x * 256 + tid;     // grid = 16 -> 4096 rows
    float y[16], x[16];
#pragma unroll
    for (int c = 0; c < 16; ++c) y[c] = Y[i * 16 + c];
#pragma unroll
    for (int c = 0; c < 16; ++c) {
        float s = y[c];
        for (int k = 0; k < c; ++k) s -= R[k * 16 + c] * x[k];
        x[c] = s / R[c * 16 + c];
    }
#pragma unroll
    for (int c = 0; c < 16; ++c) X[i * 16 + c] = x[c];
}

// ---------------------------------------------------------------------------
// 7) Rayleigh-Ritz: Jacobi eigensolve of T = X^T B X (16x16, symmetrized),
//    select the 3 algebraically largest Ritz values (descending, matches
//    evals[-1:-4:-1]). Single-thread kernel: O(10^5) flops.
// ---------------------------------------------------------------------------
__global__ void ritz_kernel(const float* __restrict__ T, float* __restrict__ Vsel,
                            float* __restrict__ lam3) {
    if (threadIdx.x != 0 || blockIdx.x != 0) return;
    float A[16][16], V[16][16];
    for (int i = 0; i < 16; ++i)
        for (int j = 0; j < 16; ++j) {
            A[i][j] = 0.5f * (T[i * 16 + j] + T[j * 16 + i]);
            V[i][j] = (i == j) ? 1.f : 0.f;
        }
    for (int sweep = 0; sweep < 30; ++sweep) {
        for (int p = 0; p < 15; ++p)
            for (int q = p + 1; q < 16; ++q) {
                float apq = A[p][q];
                if (fabsf(apq) < 1e-12f) continue;
                float theta = 0.5f * (A[q][q] - A[p][p]) / apq;
                float t = (theta >= 0.f ? 1.f : -1.f) /
                          (fabsf(theta) + sqrtf(theta * theta + 1.f));
                float c = 1.0f / sqrtf(t * t + 1.f), s = t * c;
                for (int k = 0; k < 16; ++k) {
                    float akp = A[k][p], akq = A[k][q];
                    A[k][p] = c * akp - s * akq;
                    A[k][q] = s * akp + c * akq;
                }
                for (int k = 0; k < 16; ++k) {
                    float apk = A[p][k], aqk = A[q][k];
                    A[p][k] = c * apk - s * aqk;
                    A[q][k] = s * apk + c * aqk;
                }
                for (int k = 0; k < 16; ++k) {
                    float vkp = V[k][p], vkq = V[k][q];
                    V[k][p] = c * vkp - s * vkq;
                    V[k][q] = s * vkp + c * vkq;
                }
            }
    }
    int   idx[3] = {0, 0, 0};
    float val[3] = {-3.4e38f, -3.4e38f, -3.4e38f};
    for (int j = 0; j < 16; ++j) {
        float w = A[j][j];
        if (w > val[0]) { val[2]=val[1]; idx[2]=idx[1]; val[1]=val[0]; idx[1]=idx[0]; val[0]=w; idx[0]=j; }
        else if (w > val[1]) { val[2]=val[1]; idx[2]=idx[1]; val[1]=w; idx[1]=j; }
        else if (w > val[2]) { val[2]=w; idx[2]=j; }
    }
    for (int c = 0; c < 3; ++c) {
        lam3[c] = sqrtf(fmaxf(val[c], 1e-10f));
        for (int k = 0; k < 16; ++k) Vsel[k * 3 + c] = V[k][idx[c]];
    }
}

// coords[i][c] = (X[i,:] . Vsel[:,c]) * sqrt(clip(lambda_c))
__global__ void finalcoords_kernel(const float* __restrict__ X, const float* __restrict__ Vsel,
                                   const float* __restrict__ lam3, float* __restrict__ coords) {
    int idx = blockIdx.x * blockDim.x + threadIdx.x;
    if (idx >= L * 3) return;
    int i = idx / 3, c = idx % 3;
    float s = 0.f;
#pragma unroll
    for (int k = 0; k < 16; ++k) s += X[i * 16 + k] * Vsel[k * 3 + c];
    coords[idx] = s * lam3[c];
}

// ---------------------------------------------------------------------------
// 8) Stress gradient: one block per atom k; coords staged in LDS (48KB of the
//    320KB/WGP). grad_k = 4/(W+eps) * sum_j w*(d-p)/d*(ck-cj) + chain bonds.
// ---------------------------------------------------------------------------
__global__ __launch_bounds__(256) void grad_kernel(const float* __restrict__ coords,
                                                   const float* __restrict__ predd,
                                                   const float* __restrict__ wght,
                                                   const float* __restrict__ scal,
                                                   float* __restrict__ grad) {
    __shared__ float sc[L * 3];     // 48 KB coord stage
    __shared__ float red[256 * 3];
    int k = blockIdx.x, tid = threadIdx.x;
    for (int i = tid; i < L * 3; i += 256) sc[i] = coords[i];
    __syncthreads();
    float ckx = sc[k * 3 + 0], cky = sc[k * 3 + 1], ckz = sc[k * 3 + 2];
    const float* p = predd + (size_t)k * L;
    const float* w = wght  + (size_t)k * L;
    float ax = 0.f, ay = 0.f, az = 0.f;
    for (int j = tid; j < L; j += 256) {
        float wv = w[j], pv = p[j];
        float dx = ckx - sc[j * 3 + 0];
        float dy = cky - sc[j * 3 + 1];
        float dz = ckz - sc[j * 3 + 2];
        float d = sqrtf(dx * dx + dy * dy + dz * dz + 1e-8f);
        float f = wv * (d - pv) / d;
        ax += f * dx; ay += f * dy; az += f * dz;
    }
    red[tid] = ax; red[256 + tid] = ay; red[512 + tid] = az; __syncthreads();
    for (int s = 128; s > 0; s >>= 1) {
        if (tid < s) {
            red[tid]       += red[tid + s];
            red[256 + tid] += red[256 + tid + s];
            red[512 + tid] += red[512 + tid + s];
        }
        __syncthreads();
    }
    if (tid == 0) {
        float s4 = 4.0f / (scal[1] + 1e-8f);
        float gx = red[0] * s4, gy = red[256] * s4, gz = red[512] * s4;
        const float bf = 0.1f * 2.0f / (float)(L - 1);
        if (k + 1 < L) {
            float dx = ckx - sc[(k + 1) * 3 + 0];
            float dy = cky - sc[(k + 1) * 3 + 1];
            float dz = ckz - sc[(k + 1) * 3 + 2];
            float nd = sqrtf(dx * dx + dy * dy + dz * dz + 1e-8f);
            float f = bf * (nd - 5.9f) / nd;
            gx += f * dx; gy += f * dy; gz += f * dz;
        }
        if (k >= 1) {
            float dx = ckx - sc[(k - 1) * 3 + 0];
            float dy = cky - sc[(k - 1) * 3 + 1];
            float dz = ckz - sc[(k - 1) * 3 + 2];
            float nd = sqrtf(dx * dx + dy * dy + dz * dz + 1e-8f);
            float f = bf * (nd - 5.9f) / nd;
            gx += f * dx; gy += f * dy; gz += f * dz;
        }
        grad[k * 3 + 0] = gx; grad[k * 3 + 1] = gy; grad[k * 3 + 2] = gz;
    }
}

// Adam step (bias corrections precomputed on host)
__global__ void adam_kernel(float* __restrict__ coords, const float* __restrict__ grad,
                            float* __restrict__ m, float* __restrict__ v,
                            float bc1inv, float bc2inv) {
    int i = blockIdx.x * blockDim.x + threadIdx.x;
    if (i >= L * 3) return;
    float g  = grad[i];
    float mm = 0.9f   * m[i] + 0.1f   * g;
    float vv = 0.999f * v[i] + 0.001f * g * g;
    m[i] = mm; v[i] = vv;
    coords[i] -= 0.1f * (mm * bc1inv) / (sqrtf(vv * bc2inv) + 1e-8f);
}

// ---------------------------------------------------------------------------
extern "C" void kernel_launch(void* const* d_in, const int* in_sizes, int n_in,
                              void* d_out, int out_size, void* d_ws, size_t ws_size,
                              hipStream_t stream) {
    const float* P = (const float*)d_in[0];   // pred_distances
    const float* C = (const float*)d_in[1];   // confidence
    const float* M = (const float*)d_in[2];   // mask
    float* coords  = (float*)d_out;           // 4096*3, used as live coords

    float* ws    = (float*)d_ws;
    float* predd = ws;                  // LL
    float* wght  = ws + LL;             // LL
    float* rmean = ws + 2 * LL;         // 4096
    float* rsumw = rmean + L;           // 4096
    float* X     = rsumw + L;           // 4096*16
    float* Y     = X + L * 16;          // 4096*16
    float* madam = Y + L * 16;          // 12288
    float* vadam = madam + L * 3;       // 12288
    float* grad  = vadam + L * 3;       // 12288
    float* part  = grad + L * 3;        // 16*256
    float* G     = part + 16 * 256;     // 256 (Gram / Ritz T)
    float* Vsel  = G + 256;             // 16*3
    float* lam3  = Vsel + 48;           // 3
    float* scal  = lam3 + 3;            // [0]=grandmean(D^2) [1]=sum(weight)

    // --- setup: symmetrize + centering stats ---
    sym_kernel<<<(unsigned)(LL / 256), 256, 0, stream>>>(P, C, M, predd, wght);
    rowsum_kernel<<<L, 256, 0, stream>>>(predd, wght, rmean, rsumw);
    scalars_kernel<<<1, 256, 0, stream>>>(rmean, rsumw, scal);

    // --- MDS init: 16-column subspace iteration (WMMA GEMM + WMMA Gram +
    //     CholeskyQR), then Rayleigh-Ritz for the top-3 eigenpairs ---
    initX_kernel<<<(L * 16) / 256, 256, 0, stream>>>(X);
    for (int it = 0; it < POWER_ITERS; ++it) {
        gemm_power_kernel<<<L / 16, 32, 0, stream>>>(predd, rmean, scal, X, Y);
        gram_wmma_kernel<<<16, 32, 0, stream>>>(Y, Y, part);
        gram_reduce_kernel<<<1, 256, 0, stream>>>(part, G);
        cholqr16_kernel<<<L / 256, 256, 0, stream>>>(Y, G, X);
    }
    gemm_power_kernel<<<L / 16, 32, 0, stream>>>(predd, rmean, scal, X, Y);
    gram_wmma_kernel<<<16, 32, 0, stream>>>(X, Y, part);     // T = X^T B X
    gram_reduce_kernel<<<1, 256, 0, stream>>>(part, G);
    ritz_kernel<<<1, 32, 0, stream>>>(G, Vsel, lam3);
    finalcoords_kernel<<<(L * 3 + 255) / 256, 256, 0, stream>>>(X, Vsel, lam3, coords);

    // --- Adam loop ---
    zero_kernel<<<(2 * L * 3 + 255) / 256, 256, 0, stream>>>(madam, 2 * L * 3);
    for (int it = 0; it < NUM_ITERS; ++it) {
        double t = (double)(it + 1);
        float bc1inv = (float)(1.0 / (1.0 - pow(0.9, t)));
        float bc2inv = (float)(1.0 / (1.0 - pow(0.999, t)));
        grad_kernel<<<L, 256, 0, stream>>>(coords, predd, wght, scal, grad);
        adam_kernel<<<(L * 3 + 255) / 256, 256, 0, stream>>>(coords, grad, madam, vadam,
                                                             bc1inv, bc2inv);
    }
}